// WeightedAveragePrevEmbeddingsLM_82214263980435
// MI455X (gfx1250) — compile-verified
//
#include <hip/hip_runtime.h>
#include <hip/hip_bf16.h>
#include <math.h>

// Shapes
#define VOCAB 32000
#define EMB   16
#define BS    4
#define MCW   2048
#define NTOK  (BS * MCW)          // 8192 rows

typedef float v2f __attribute__((ext_vector_type(2)));
typedef float v4f __attribute__((ext_vector_type(4)));
typedef float v8f __attribute__((ext_vector_type(8)));

__device__ __forceinline__ float dot4(v4f a, v4f b) {
    return fmaf(a.x, b.x, fmaf(a.y, b.y, fmaf(a.z, b.z, a.w * b.w)));
}

// ---------------------------------------------------------------------------
// Kernel 1: gather embeddings  embs[t][e] = table[x[t]][e]
// ---------------------------------------------------------------------------
__global__ void k_gather(const int* __restrict__ x,
                         const float* __restrict__ table,
                         float* __restrict__ embs) {
    int gid = blockIdx.x * blockDim.x + threadIdx.x;   // 131072 threads
    int t = gid >> 4;
    int e = gid & 15;
    int tok = x[t];
    embs[gid] = table[(size_t)tok * EMB + e];
}

// ---------------------------------------------------------------------------
// Kernel 2: per-COLUMN softmax stats (softmax over query axis, faithful).
// One wave32 per (b, k). Valid entries: q in [k, MCW) with s(q,k) != 0.
// Online (m, z) accumulation + butterfly combine.
// ---------------------------------------------------------------------------
__global__ void k_colstats(const float* __restrict__ embs,
                           float* __restrict__ cmax,
                           float* __restrict__ csum) {
    int wave = blockIdx.x * (blockDim.x >> 5) + (threadIdx.x >> 5);
    int lane = threadIdx.x & 31;
    int b = wave >> 11;          // / MCW
    int k = wave & (MCW - 1);
    const float* eb = embs + (size_t)b * MCW * EMB;

    const float* ek = eb + (size_t)k * EMB;
    v4f k0 = *(const v4f*)(ek + 0);
    v4f k1 = *(const v4f*)(ek + 4);
    v4f k2 = *(const v4f*)(ek + 8);
    v4f k3 = *(const v4f*)(ek + 12);

    float m = -__builtin_inff();
    float z = 0.0f;
    for (int q = k + lane; q < MCW; q += 32) {
        const float* eq = eb + (size_t)q * EMB;
        float s = dot4(*(const v4f*)(eq + 0),  k0)
                + dot4(*(const v4f*)(eq + 4),  k1)
                + dot4(*(const v4f*)(eq + 8),  k2)
                + dot4(*(const v4f*)(eq + 12), k3);
        if (s != 0.0f) {                  // exact zeros -> -inf -> excluded
            float mn = fmaxf(m, s);
            z = z * __expf(m - mn) + __expf(s - mn);   // z==0 when m==-inf
            m = mn;
        }
    }
    // butterfly combine across the wave32
    #pragma unroll
    for (int off = 16; off >= 1; off >>= 1) {
        float m2 = __shfl_xor(m, off, 32);
        float z2 = __shfl_xor(z, off, 32);
        float mn = fmaxf(m, m2);
        float za = (m  > -__builtin_inff()) ? z  * __expf(m  - mn) : 0.0f;
        float zb = (m2 > -__builtin_inff()) ? z2 * __expf(m2 - mn) : 0.0f;
        z = za + zb;
        m = mn;
    }
    if (lane == 0) {
        cmax[wave] = m;
        csum[wave] = z;
    }
}

// ---------------------------------------------------------------------------
// Kernel 3: out[b][q][:] = sum_{k<=q, s!=0} exp(s - m_k)/Z_k * embs[b][k][:]
// One wave32 per (b, q); lanes stride over k; butterfly-sum 16 accumulators.
// ---------------------------------------------------------------------------
__global__ void k_attnout(const float* __restrict__ embs,
                          const float* __restrict__ cmax,
                          const float* __restrict__ csum,
                          float* __restrict__ outa) {
    int wave = blockIdx.x * (blockDim.x >> 5) + (threadIdx.x >> 5);
    int lane = threadIdx.x & 31;
    int b = wave >> 11;
    int q = wave & (MCW - 1);
    const float* eb = embs + (size_t)b * MCW * EMB;
    const float* cm = cmax + (size_t)b * MCW;
    const float* cz = csum + (size_t)b * MCW;

    const float* eq = eb + (size_t)q * EMB;
    v4f q0 = *(const v4f*)(eq + 0);
    v4f q1 = *(const v4f*)(eq + 4);
    v4f q2 = *(const v4f*)(eq + 8);
    v4f q3 = *(const v4f*)(eq + 12);

    v4f a0 = {0.f,0.f,0.f,0.f}, a1 = a0, a2 = a0, a3 = a0;
    for (int k = lane; k <= q; k += 32) {
        const float* ek = eb + (size_t)k * EMB;
        v4f e0 = *(const v4f*)(ek + 0);
        v4f e1 = *(const v4f*)(ek + 4);
        v4f e2 = *(const v4f*)(ek + 8);
        v4f e3 = *(const v4f*)(ek + 12);
        float s = dot4(q0, e0) + dot4(q1, e1) + dot4(q2, e2) + dot4(q3, e3);
        if (s != 0.0f) {
            float w = __expf(s - cm[k]) / cz[k];
            a0 += w * e0; a1 += w * e1; a2 += w * e2; a3 += w * e3;
        }
    }
    // reduce all 16 components across the wave
    #pragma unroll
    for (int off = 16; off >= 1; off >>= 1) {
        a0.x += __shfl_xor(a0.x, off, 32); a0.y += __shfl_xor(a0.y, off, 32);
        a0.z += __shfl_xor(a0.z, off, 32); a0.w += __shfl_xor(a0.w, off, 32);
        a1.x += __shfl_xor(a1.x, off, 32); a1.y += __shfl_xor(a1.y, off, 32);
        a1.z += __shfl_xor(a1.z, off, 32); a1.w += __shfl_xor(a1.w, off, 32);
        a2.x += __shfl_xor(a2.x, off, 32); a2.y += __shfl_xor(a2.y, off, 32);
        a2.z += __shfl_xor(a2.z, off, 32); a2.w += __shfl_xor(a2.w, off, 32);
        a3.x += __shfl_xor(a3.x, off, 32); a3.y += __shfl_xor(a3.y, off, 32);
        a3.z += __shfl_xor(a3.z, off, 32); a3.w += __shfl_xor(a3.w, off, 32);
    }
    if (lane == 0) {
        float* o = outa + (size_t)wave * EMB;
        *(v4f*)(o + 0)  = a0;
        *(v4f*)(o + 4)  = a1;
        *(v4f*)(o + 8)  = a2;
        *(v4f*)(o + 12) = a3;
    }
}

// ---------------------------------------------------------------------------
// Kernel 4: logits = outa(8192x16) @ W^T(16x32000) + b  via WMMA f32 16x16x4.
// One wave per 16-row x 64-col strip (4 tiles of 16x16, A fragments reused).
// A 16x4 f32 fragment: lanes 0-15 hold M=lane, K={0,1}; lanes 16-31 K={2,3}.
// B 4x16 fragment (mirrored): lanes 0-15 hold N=lane, K={0,1}; 16-31 K={2,3}.
// C/D 16x16 f32: VGPR i holds row (i + 8*(lane/16)), col (lane%16).
// EXEC is all-1s: every wave is fully populated (8192 and 32000 are /16).
// Output (1.05 GB >> 192 MB L2) is streamed with non-temporal stores so the
// reused operands (W: 2 MB, A: 512 KB) stay L2-resident.
// ---------------------------------------------------------------------------
#define COL_TILES_PER_WAVE 4
#define N_COL_GROUPS (VOCAB / (16 * COL_TILES_PER_WAVE))   // 500
#define N_ROW_TILES  (NTOK / 16)                           // 512

__global__ void k_logits(const float* __restrict__ outa,
                         const float* __restrict__ W,
                         const float* __restrict__ bias,
                         float* __restrict__ logits) {
    int wave = blockIdx.x * (blockDim.x >> 5) + (threadIdx.x >> 5);
    int lane = threadIdx.x & 31;
    int colGroup = wave % N_COL_GROUPS;
    int rowTile  = wave / N_COL_GROUPS;
    int r0 = rowTile * 16;
    int half = lane >> 4;      // 0: K pair {0,1}; 1: K pair {2,3}
    int lm   = lane & 15;

    // A fragments for the 4 chained K-chunks (K = kc*4 .. kc*4+3)
    v2f afr[4];
    #pragma unroll
    for (int kc = 0; kc < 4; ++kc)
        afr[kc] = *(const v2f*)(outa + (size_t)(r0 + lm) * EMB + kc * 4 + 2 * half);

    #pragma unroll
    for (int t = 0; t < COL_TILES_PER_WAVE; ++t) {
        int n0 = (colGroup * COL_TILES_PER_WAVE + t) * 16;
        v8f c = {0.f,0.f,0.f,0.f,0.f,0.f,0.f,0.f};
        #pragma unroll
        for (int kc = 0; kc < 4; ++kc) {
            // B[K][N] = W[N][K]: lane lm gets W[n0+lm][kc*4 + 2*half .. +1]
            v2f bfr = *(const v2f*)(W + (size_t)(n0 + lm) * EMB + kc * 4 + 2 * half);
            c = __builtin_amdgcn_wmma_f32_16x16x4_f32(
                    false, afr[kc], false, bfr, (short)0, c, false, false);
        }
        float bv = bias[n0 + lm];
        float* dst = logits + (size_t)(r0 + 8 * half) * VOCAB + n0 + lm;
        #pragma unroll
        for (int i = 0; i < 8; ++i) {
            // streaming store: bypass/NT hint — output is written exactly once
            __builtin_nontemporal_store(c[i] + bv, dst + (size_t)i * VOCAB);
        }
    }
}

// ---------------------------------------------------------------------------
extern "C" void kernel_launch(void* const* d_in, const int* in_sizes, int n_in,
                              void* d_out, int out_size, void* d_ws, size_t ws_size,
                              hipStream_t stream) {
    const int*   x     = (const int*)d_in[0];
    const float* table = (const float*)d_in[1];
    const float* W     = (const float*)d_in[2];
    const float* bias  = (const float*)d_in[3];
    float*       out   = (float*)d_out;

    float* ws   = (float*)d_ws;
    float* embs = ws;                        // NTOK*EMB = 131072 floats
    float* cmax = embs + (size_t)NTOK * EMB; // 8192 floats
    float* csum = cmax + NTOK;               // 8192 floats
    float* outa = csum + NTOK;               // 131072 floats

    // 1) gather: 131072 threads
    k_gather<<<(NTOK * EMB) / 256, 256, 0, stream>>>(x, table, embs);

    // 2) column stats: 8192 waves, 8 waves/block
    k_colstats<<<NTOK / 8, 256, 0, stream>>>(embs, cmax, csum);

    // 3) attention output rows: 8192 waves
    k_attnout<<<NTOK / 8, 256, 0, stream>>>(embs, cmax, csum, outa);

    // 4) WMMA projection: 512*500 = 256000 waves, 8 waves/block
    k_logits<<<(N_ROW_TILES * N_COL_GROUPS) / 8, 256, 0, stream>>>(outa, W, bias, out);
}